// CliffordLinear_74835510165750
// MI455X (gfx1250) — compile-verified
//
#include <hip/hip_runtime.h>
#include <hip/hip_bf16.h>

#define DIM   64
#define NPAIR 2016

typedef __attribute__((ext_vector_type(2))) float v2f;
typedef __attribute__((ext_vector_type(8))) float v8f;

// ---------------------------------------------------------------------------
// Kernel 1: collapse the 2016 sequential Givens rotations into one 64x64
// matrix R (out_row = R @ x_row). Thread t owns column t of R; each rotation
// step only mixes rows i and j of that column, so threads are independent.
// R is written row-major to the workspace: R[row*64 + col].
// ---------------------------------------------------------------------------
__global__ void __launch_bounds__(DIM) build_R_kernel(const float* __restrict__ coeffs,
                                                      float* __restrict__ R) {
    __shared__ float Rs[DIM][DIM];
    const int t = threadIdx.x;            // column owned by this thread
    for (int r = 0; r < DIM; ++r)
        Rs[r][t] = (r == t) ? 1.0f : 0.0f;

    int p = 0;
    for (int i = 0; i < DIM - 1; ++i) {
        for (int j = i + 1; j < DIM; ++j) {
            const float th = coeffs[p++];
            float s, c;
            __sincosf(th, &s, &c);
            const float ri = Rs[i][t];
            const float rj = Rs[j][t];
            Rs[i][t] = c * ri - s * rj;
            Rs[j][t] = s * ri + c * rj;
        }
    }
    // row-major store, coalesced across threads
    for (int r = 0; r < DIM; ++r)
        R[r * DIM + t] = Rs[r][t];
}

// ---------------------------------------------------------------------------
// Kernel 2: out[n, d] = sum_k x[n, k] * R[d, k] + bias[d]
// 256 threads = 8 waves; each wave computes a 16-row x 64-col output tile with
// V_WMMA_F32_16X16X4_F32 (fp32 in/out, no precision loss; HBM-bound anyway).
//
// f32 WMMA operand layouts (CDNA5 ISA 7.12.2):
//   A 16x4 : lane m = lane&15 is row M; half = lane>>4; VGPR pair holds
//            K = k0+2*half, k0+2*half+1  -> contiguous float2 of x row M.
//   B 4x16 : lane n = lane&15 is col N; VGPR pair holds K = k0+2*half..+1
//            B[k][n] = R[n][k]          -> contiguous float2 of R row n.
//   C/D    : 8 VGPRs; col = lane&15 (same for all 8), row = v + 8*half.
// ---------------------------------------------------------------------------
__global__ void __launch_bounds__(256) clifford_gemm_kernel(
        const float* __restrict__ x,
        const float* __restrict__ R,
        const float* __restrict__ bias,
        float* __restrict__ out) {
    __shared__ float Rs[DIM * DIM];

    const int tid = threadIdx.x;
    // Stage R (16 KB) into LDS with coalesced b128 loads.
    {
        const float4* src = (const float4*)R;
        float4*       dst = (float4*)Rs;
        #pragma unroll
        for (int i = 0; i < (DIM * DIM / 4) / 256; ++i)
            dst[tid + i * 256] = src[tid + i * 256];
    }
    __syncthreads();

    const int wave = tid >> 5;
    const int lane = tid & 31;
    const int m    = lane & 15;   // A row / B col / D col within tile
    const int half = lane >> 4;

    const long rowbase = ((long)blockIdx.x * 8 + wave) * 16;
    const float* xrow  = x + (rowbase + m) * DIM;

    // Preload all 16 A fragments (the full K=64 strip of this lane's x row).
    v2f A[16];
    #pragma unroll
    for (int t = 0; t < 16; ++t)
        A[t] = *(const v2f*)(xrow + 4 * t + 2 * half);

    #pragma unroll
    for (int ct = 0; ct < 4; ++ct) {
        const float bv = bias[ct * 16 + m];        // D col is lane-constant
        v8f acc = {bv, bv, bv, bv, bv, bv, bv, bv}; // fold bias into C

        const float* brow = &Rs[(ct * 16 + m) * DIM];
        #pragma unroll
        for (int t = 0; t < 16; ++t) {
            v2f b = *(const v2f*)(brow + 4 * t + 2 * half);
            acc = __builtin_amdgcn_wmma_f32_16x16x4_f32(
                /*neg_a=*/false, A[t], /*neg_b=*/false, b,
                /*c_mod=*/(short)0, acc, /*reuse_a=*/false, /*reuse_b=*/false);
        }

        float* orow = out + rowbase * DIM + ct * 16 + m;
        #pragma unroll
        for (int v = 0; v < 8; ++v)
            orow[(long)(v + 8 * half) * DIM] = acc[v];
    }
}

extern "C" void kernel_launch(void* const* d_in, const int* in_sizes, int n_in,
                              void* d_out, int out_size, void* d_ws, size_t ws_size,
                              hipStream_t stream) {
    const float* x      = (const float*)d_in[0];
    const float* coeffs = (const float*)d_in[1];
    const float* bias   = (const float*)d_in[2];
    float*       out    = (float*)d_out;
    float*       R      = (float*)d_ws;           // 64*64*4 = 16 KB scratch

    const long N = (long)in_sizes[0] / DIM;       // 524288 rows

    build_R_kernel<<<1, DIM, 0, stream>>>(coeffs, R);

    // 8 waves/block * 16 rows/wave = 128 rows per block
    const int blocks = (int)(N / 128);
    clifford_gemm_kernel<<<blocks, 256, 0, stream>>>(x, R, bias, out);
}